// OTTT_63273458205169
// MI455X (gfx1250) — compile-verified
//
#include <hip/hip_runtime.h>

// ---- problem constants (match reference) ----
#define IN_DIM   8192
#define OUT_DIM  8192
#define SIG_TAU  0.88079707797788231f   // sigmoid(2.0)
#define V_TH     1.0f

// ---- tiling ----
#define WAVES_PER_BLOCK 4
#define BLOCK_THREADS   (WAVES_PER_BLOCK * 32)            // 128 threads = 4 wave32
#define TILES_PER_WAVE  4                                 // 4x 16-col WMMA tiles
#define COLS_PER_WAVE   (TILES_PER_WAVE * 16)             // 64
#define COLS_PER_BLOCK  (WAVES_PER_BLOCK * COLS_PER_WAVE) // 256
#define MAX_KSPLITS     16

typedef __attribute__((ext_vector_type(2))) float v2f;
typedef __attribute__((ext_vector_type(8))) float v8f;

// Pass 1: split-K GEMV partials via V_WMMA_F32_16X16X4_F32.
// A (16x4) = x[k..k+3] replicated across the 16 rows (one b64 load, shared by
// all 4 column tiles) -> every row of D is the GEMV result; harvest row 0.
__global__ __launch_bounds__(BLOCK_THREADS)
void ottt_gemv_wmma(const float* __restrict__ W,
                    const float* __restrict__ x,
                    float* __restrict__ partial,
                    int kchunk) {
    const int lane  = threadIdx.x & 31;
    const int wave  = threadIdx.x >> 5;
    const int j0    = blockIdx.x * COLS_PER_BLOCK + wave * COLS_PER_WAVE;
    const int kbase = blockIdx.y * kchunk;

    // ISA 7.12.2 fp32 layouts: lanes 0..15 carry K=0/1, lanes 16..31 carry K=2/3.
    const int krow0 = (lane < 16) ? 0 : 2;  // K offset for this half-wave
    const int n     = lane & 15;            // column within a 16-wide tile

    const float* __restrict__ Wbase = W + (size_t)j0 + (size_t)n;

    v8f acc0 = {}, acc1 = {}, acc2 = {}, acc3 = {};

    const int kend = kbase + kchunk;
    #pragma unroll 2
    for (int k = kbase; k < kend; k += 4) {
        // A fragment: x broadcast down the rows; contiguous pair -> b64 load.
        const v2f a = *(const v2f*)(x + k + krow0);
        // B fragments: W rows k+krow0, k+krow0+1 ; 4 adjacent 16-col tiles.
        const float* __restrict__ r0 = Wbase + (size_t)(k + krow0) * OUT_DIM;
        const float* __restrict__ r1 = r0 + OUT_DIM;
        v2f b0 = { r0[ 0], r1[ 0] };
        v2f b1 = { r0[16], r1[16] };
        v2f b2 = { r0[32], r1[32] };
        v2f b3 = { r0[48], r1[48] };
        // (neg_a, A, neg_b, B, c_mod, C, reuse_a, reuse_b)
        acc0 = __builtin_amdgcn_wmma_f32_16x16x4_f32(false, a, false, b0, (short)0, acc0, false, false);
        acc1 = __builtin_amdgcn_wmma_f32_16x16x4_f32(false, a, false, b1, (short)0, acc1, false, false);
        acc2 = __builtin_amdgcn_wmma_f32_16x16x4_f32(false, a, false, b2, (short)0, acc2, false, false);
        acc3 = __builtin_amdgcn_wmma_f32_16x16x4_f32(false, a, false, b3, (short)0, acc3, false, false);
    }

    // D row 0 lives in acc[0] on lanes 0..15 (all rows identical by construction).
    if (lane < 16) {
        float* __restrict__ p = partial + (size_t)blockIdx.y * OUT_DIM + (size_t)j0 + n;
        p[ 0] = acc0[0];
        p[16] = acc1[0];
        p[32] = acc2[0];
        p[48] = acc3[0];
    }
}

// Pass 2: deterministic fixed-order partial reduction + LIF epilogue + trace.
// d_out layout: [0,8192) s ; [8192,16384) u_new ; [16384,24576) a_hat_new.
__global__ __launch_bounds__(256)
void ottt_finalize(const float* __restrict__ partial,
                   const float* __restrict__ b,
                   const float* __restrict__ u,
                   const float* __restrict__ a_hat,
                   const float* __restrict__ x,
                   float* __restrict__ out,
                   int ksplits) {
    const int j = blockIdx.x * blockDim.x + threadIdx.x;
    if (j >= OUT_DIM) return;

    float acc = 0.0f;
    for (int s = 0; s < ksplits; ++s)        // fixed order -> deterministic
        acc += partial[(size_t)s * OUT_DIM + j];

    const float u_pre = SIG_TAU * u[j] + acc + b[j];
    const float spike = (u_pre >= V_TH) ? 1.0f : 0.0f;

    out[j]               = spike;                       // s
    out[OUT_DIM + j]     = u_pre - spike * V_TH;        // u_new (soft reset)
    out[2 * OUT_DIM + j] = SIG_TAU * a_hat[j] + x[j];   // a_hat_new
}

extern "C" void kernel_launch(void* const* d_in, const int* in_sizes, int n_in,
                              void* d_out, int out_size, void* d_ws, size_t ws_size,
                              hipStream_t stream) {
    (void)in_sizes; (void)n_in; (void)out_size;
    const float* W     = (const float*)d_in[0];
    const float* bvec  = (const float*)d_in[1];
    const float* u     = (const float*)d_in[2];
    const float* a_hat = (const float*)d_in[3];
    const float* x     = (const float*)d_in[4];
    float* out         = (float*)d_out;
    float* partial     = (float*)d_ws;

    // Largest power-of-two split count (<=16) whose partial buffer fits in d_ws.
    int ksplits = MAX_KSPLITS;
    while (ksplits > 1 && (size_t)ksplits * OUT_DIM * sizeof(float) > ws_size)
        ksplits >>= 1;
    const int kchunk = IN_DIM / ksplits;     // multiple of 4 for all splits

    dim3 grid1(OUT_DIM / COLS_PER_BLOCK, ksplits);   // (32, ksplits)
    ottt_gemv_wmma<<<grid1, BLOCK_THREADS, 0, stream>>>(W, x, partial, kchunk);

    dim3 grid2((OUT_DIM + 255) / 256);
    ottt_finalize<<<grid2, 256, 0, stream>>>(partial, bvec, u, a_hat, x, out, ksplits);
}